// SelfAttention_27848567947507
// MI455X (gfx1250) — compile-verified
//
#include <hip/hip_runtime.h>

// ---------------------------------------------------------------------------
// MI455X (gfx1250) local self-attention (SASA, 7x7, 8 heads x d=8, C=64).
//
// Roofline: ~1.4 GFLOP total vs ~17 MB HBM traffic (intermediates are
// L2-resident: 25 MB << 192 MB L2)  ->  memory-bound, ~1 us floor at
// 23.3 TB/s. So: keep everything f32 (exact vs reference), use the native
// f32 WMMA (V_WMMA_F32_16X16X4_F32) for the only real GEMMs (QKV
// projections), and use CDNA5's 320KB-per-WGP LDS to stage the haloed
// K/V tiles for the gather-heavy attention stage.
// ---------------------------------------------------------------------------

typedef float v2f __attribute__((ext_vector_type(2)));
typedef float v8f __attribute__((ext_vector_type(8)));

#define BATCH   8
#define IMG     64
#define CIN     64
#define NPIX    (BATCH * IMG * IMG)   // 32768
#define NH      8
#define DHEAD   8
#define KS      7
#define KK      (KS * KS)             // 49
#define PAD     3

// ---------------------------------------------------------------------------
// Kernel 1: Qp/Kp/Vp = x @ {W_Q,W_K,W_V}^T   (per-pixel 64->64 projections).
// One wave computes one 16-pixel x 16-channel tile for Q, K and V at once
// (A-operand reuse). f32 WMMA 16x16x4, K-loop over 64 input channels.
//
// Per ISA layouts (wave32):
//   A 16x4 f32 : lanes 0-15 hold rows M=0..15 with VGPR0=K0,VGPR1=K1;
//                lanes 16-31 hold K2,K3            -> one b64 load / lane.
//   B 4x16 f32 : lanes 0-15 hold N=0..15 with VGPR0=Krow0,VGPR1=Krow1;
//                lanes 16-31 hold Krow2,Krow3      -> one b64 load / lane.
//   C/D 16x16  : VGPR j, lanes 0-15 -> M=j ; lanes 16-31 -> M=j+8 ; N=lane%16.
// ---------------------------------------------------------------------------
__global__ __launch_bounds__(256) void qkv_proj_kernel(
    const float* __restrict__ x,
    const float* __restrict__ wq,   // [NH, DHEAD, CIN] == [64, 64] row-major
    const float* __restrict__ wk,
    const float* __restrict__ wv,
    float* __restrict__ Qp,         // [NPIX, 64]
    float* __restrict__ Kp,
    float* __restrict__ Vp)
{
    const int wave   = threadIdx.x >> 5;
    const int lane   = threadIdx.x & 31;
    const int tileId = blockIdx.x * 8 + wave;   // 8192 tiles, grid sized exactly
    const int pTile  = tileId >> 2;             // 0..2047 : 16-pixel tile
    const int nTile  = tileId & 3;              // 0..3    : 16-channel tile

    const int half = lane >> 4;                 // 0 | 1
    const int r    = lane & 15;
    const int pix0 = pTile * 16;
    const int nb   = nTile * 16;

    v8f c0 = {}; v8f c1 = {}; v8f c2 = {};

    // A: x[pix0 + r, k0 + 2*half .. +1]     B: W[nb + r, k0 + 2*half .. +1]
    const float* aptr = x  + (size_t)(pix0 + r) * CIN + 2 * half;
    const float* bq   = wq + (size_t)(nb + r)   * CIN + 2 * half;
    const float* bk   = wk + (size_t)(nb + r)   * CIN + 2 * half;
    const float* bv   = wv + (size_t)(nb + r)   * CIN + 2 * half;

    #pragma unroll
    for (int k0 = 0; k0 < CIN; k0 += 4) {
        v2f a   = *(const v2f*)(aptr + k0);
        v2f bbq = *(const v2f*)(bq + k0);
        v2f bbk = *(const v2f*)(bk + k0);
        v2f bbv = *(const v2f*)(bv + k0);
        c0 = __builtin_amdgcn_wmma_f32_16x16x4_f32(false, a, false, bbq, (short)0, c0, false, false);
        c1 = __builtin_amdgcn_wmma_f32_16x16x4_f32(false, a, false, bbk, (short)0, c1, false, false);
        c2 = __builtin_amdgcn_wmma_f32_16x16x4_f32(false, a, false, bbv, (short)0, c2, false, false);
    }

    #pragma unroll
    for (int j = 0; j < 8; ++j) {
        const size_t row = (size_t)(pix0 + j + 8 * half) * 64 + nb + r;
        Qp[row] = c0[j];
        Kp[row] = c1[j];
        Vp[row] = c2[j];
    }
}

// ---------------------------------------------------------------------------
// Kernel 2: local attention. Block = 8x16 spatial tile (128 threads, one
// pixel per thread), all 8 heads. The haloed 14x22 K and V tiles live in
// LDS (CDNA5: 320KB/WGP). Per-pixel LDS stride = 65 floats so that
// bank(pix,c) = (pix + c) mod 64 -> conflict-free scalar reads for a
// wave's 16-consecutive-x + next-row lane pattern.
//
// Out-of-image neighbors: K=V=0 (zero-padded x, linear projection), but the
// rel-pos logit still applies and they stay in the softmax — matches ref.
// dot_QK + rel  ==  q . (K_neighbor + rel_pos[h,nb]).
// ---------------------------------------------------------------------------
#define TSY  8
#define TSX  16
#define ESY  (TSY + 2 * PAD)    // 14
#define ESX  (TSX + 2 * PAD)    // 22
#define NHALO (ESY * ESX)       // 308
#define PSTR 65                 // LDS floats per halo pixel (bank-staggered)

__global__ __launch_bounds__(128) void local_attn_kernel(
    const float* __restrict__ Qp,
    const float* __restrict__ Kp,
    const float* __restrict__ Vp,
    const float* __restrict__ relH,   // [NH, 7, 1, 4]
    const float* __restrict__ relW,   // [NH, 1, 7, 4]
    float* __restrict__ out)          // [B, 64, 64, 64]
{
    __shared__ float sK[NHALO * PSTR];          //  80,080 B
    __shared__ float sV[NHALO * PSTR];          //  80,080 B
    __shared__ float sRel[NH * KK * DHEAD];     //  12,544 B   total ~172.7 KB

    const int tid  = threadIdx.x;
    const int bIdx = blockIdx.x;                // 256 blocks: b*32 + ty*4 + tx
    const int b    = bIdx >> 5;
    const int rem  = bIdx & 31;
    const int tyb  = (rem >> 2) * TSY;          // tile origin y
    const int txb  = (rem & 3)  * TSX;          // tile origin x

    // --- stage rel_pos[h, nb, d] = concat(Rel_H[h, nb/7, 0, :], Rel_W[h, 0, nb%7, :])
    for (int i = tid; i < NH * KK * DHEAD; i += 128) {
        int h  = i / (KK * DHEAD);
        int rr = i - h * (KK * DHEAD);
        int nb = rr >> 3;
        int d  = rr & 7;
        int ki = nb / KS;
        int kj = nb - ki * KS;
        float v = (d < 4) ? relH[h * 28 + ki * 4 + d]
                          : relW[h * 28 + kj * 4 + (d - 4)];
        sRel[i] = v;
    }

    // --- stage haloed K and V tiles (zero-fill outside the image)
    for (int i = tid; i < NHALO * 16; i += 128) {
        int p  = i >> 4;              // halo pixel 0..307
        int c4 = (i & 15) * 4;        // channel chunk
        int hy = p / ESX;
        int hx = p - hy * ESX;
        int gy = tyb + hy - PAD;
        int gx = txb + hx - PAD;
        float4 kq = make_float4(0.f, 0.f, 0.f, 0.f);
        float4 vq = make_float4(0.f, 0.f, 0.f, 0.f);
        if (gy >= 0 && gy < IMG && gx >= 0 && gx < IMG) {
            size_t g = ((size_t)((b * IMG + gy) * IMG + gx)) * 64 + c4;
            kq = *(const float4*)(Kp + g);
            vq = *(const float4*)(Vp + g);
        }
        int base = p * PSTR + c4;
        sK[base + 0] = kq.x; sK[base + 1] = kq.y; sK[base + 2] = kq.z; sK[base + 3] = kq.w;
        sV[base + 0] = vq.x; sV[base + 1] = vq.y; sV[base + 2] = vq.z; sV[base + 3] = vq.w;
    }
    __syncthreads();

    const int ty = tid >> 4;          // 0..7
    const int tx = tid & 15;          // 0..15
    const int gy = tyb + ty;
    const int gx = txb + tx;
    const size_t pix = (size_t)((b * IMG + gy) * IMG + gx);
    const int hbase = ty * ESX + tx;  // halo coord of neighbor (0,0)

    for (int h = 0; h < NH; ++h) {    // keep rolled: per-head body is large
        // q[0..7] from L2-resident Qp
        float q[8];
        float4 qa = *(const float4*)(Qp + pix * 64 + h * 8);
        float4 qb = *(const float4*)(Qp + pix * 64 + h * 8 + 4);
        q[0] = qa.x; q[1] = qa.y; q[2] = qa.z; q[3] = qa.w;
        q[4] = qb.x; q[5] = qb.y; q[6] = qb.z; q[7] = qb.w;

        // scores = (q . (K_nb + rel_nb)) / sqrt(8), kept in VGPRs
        float s[KK];
        float mx = -3.402823466e38f;
        #pragma unroll
        for (int nb = 0; nb < KK; ++nb) {
            const int i  = nb / KS;
            const int j  = nb - i * KS;
            const int hp = hbase + i * ESX + j;
            const float* kp = &sK[hp * PSTR + h * 8];
            const float* rp = &sRel[(h * KK + nb) * 8];
            float acc = 0.f;
            #pragma unroll
            for (int d = 0; d < 8; ++d)
                acc = fmaf(q[d], kp[d] + rp[d], acc);
            acc *= 0.3535533905932738f;   // 1/sqrt(8)
            s[nb] = acc;
            mx = fmaxf(mx, acc);
        }

        float sum = 0.f;
        #pragma unroll
        for (int nb = 0; nb < KK; ++nb) {
            float e = __expf(s[nb] - mx);
            s[nb] = e;
            sum += e;
        }
        const float inv = 1.0f / sum;

        float o[8] = {0.f, 0.f, 0.f, 0.f, 0.f, 0.f, 0.f, 0.f};
        #pragma unroll
        for (int nb = 0; nb < KK; ++nb) {
            const int i  = nb / KS;
            const int j  = nb - i * KS;
            const int hp = hbase + i * ESX + j;
            const float* vp = &sV[hp * PSTR + h * 8];
            const float w = s[nb];
            #pragma unroll
            for (int d = 0; d < 8; ++d)
                o[d] = fmaf(w, vp[d], o[d]);
        }

        #pragma unroll
        for (int d = 0; d < 8; ++d)
            out[pix * 64 + h * 8 + d] = o[d] * inv;
    }
}

// ---------------------------------------------------------------------------
extern "C" void kernel_launch(void* const* d_in, const int* in_sizes, int n_in,
                              void* d_out, int out_size, void* d_ws, size_t ws_size,
                              hipStream_t stream) {
    const float* x  = (const float*)d_in[0];
    const float* wq = (const float*)d_in[1];
    const float* wk = (const float*)d_in[2];
    const float* wv = (const float*)d_in[3];
    const float* rh = (const float*)d_in[4];
    const float* rw = (const float*)d_in[5];
    float* outp = (float*)d_out;

    float* Qp = (float*)d_ws;               // 3 x 32768 x 64 f32 = 25.2 MB scratch
    float* Kp = Qp + (size_t)NPIX * 64;
    float* Vp = Kp + (size_t)NPIX * 64;

    // 8192 wave-tiles, 8 waves (256 threads) per block
    qkv_proj_kernel<<<1024, 256, 0, stream>>>(x, wq, wk, wv, Qp, Kp, Vp);
    // 8 batches x 8 y-tiles x 4 x-tiles = 256 blocks of 128 threads
    local_attn_kernel<<<256, 128, 0, stream>>>(Qp, Kp, Vp, rh, rw, outp);
}